// GNN_3770981286765
// MI455X (gfx1250) — compile-verified
//
#include <hip/hip_runtime.h>

typedef float v2f __attribute__((ext_vector_type(2)));
typedef float v8f __attribute__((ext_vector_type(8)));

#define D 128  // feature dim (D_IN == D_HID == 128)

// ---------------- degree from in-edges (self loop added analytically) ---------
__global__ void k_deg_edges(const int* __restrict__ ei, int E, float* __restrict__ deg) {
    int e = blockIdx.x * blockDim.x + threadIdx.x;
    if (e < E) atomicAdd(&deg[ei[E + e]], 1.0f);   // dst row of edge_index
}

__global__ void k_dinv(const float* __restrict__ deg, float* __restrict__ dinv, int N) {
    int n = blockIdx.x * blockDim.x + threadIdx.x;
    if (n < N) dinv[n] = rsqrtf(deg[n] + 1.0f);    // +1 self loop; always > 0
}

// ---------------- padding mask on INPUT features + per-graph node counts ------
__global__ void k_mask_cnt(const float* __restrict__ x, const int* __restrict__ batch,
                           float* __restrict__ mask, float* __restrict__ cnt, int N) {
    int n = blockIdx.x * blockDim.x + threadIdx.x;
    if (n >= N) return;
    const float* row = x + (size_t)n * D;
    float s = 0.f;
    for (int f = 0; f < D; ++f) s += row[f];
    float m = (s != 0.f) ? 1.f : 0.f;
    mask[n] = m;
    if (m != 0.f) atomicAdd(&cnt[batch[n]], 1.0f);
}

// ---------------- H = X * W via V_WMMA_F32_16X16X4_F32 ------------------------
// One wave per 16x16 output tile; 8 waves/block -> one 16-row x 128-col strip.
// A 16x4 layout (ISA 7.12.2): lane = M (0..15 twice); VGPR0 = {K=k0, K=k0+2},
// VGPR1 = {K=k0+1, K=k0+3}  => lane-half h picks K-pair k0+2h, k0+2h+1.
// B 4x16: rows striped across lanes within a VGPR, same half-split on K.
// C/D 16x16: VGPR v -> row = v + 8*half, col = lane%16.
__global__ __launch_bounds__(256)
void k_gemm_wmma_f32(const float* __restrict__ X, const float* __restrict__ W,
                     float* __restrict__ H) {
    const int wave = threadIdx.x >> 5;
    const int lane = threadIdx.x & 31;
    const int half = lane >> 4;
    const int l16  = lane & 15;
    const int row0 = blockIdx.x * 16;   // N % 16 == 0 (50000 = 3125*16)
    const int col0 = wave * 16;

    const float* xrow = X + (size_t)(row0 + l16) * D;
    v8f c = {0.f, 0.f, 0.f, 0.f, 0.f, 0.f, 0.f, 0.f};

    for (int k0 = 0; k0 < D; k0 += 4) {
        const int ka = k0 + 2 * half;
        v2f a, b;
        a.x = xrow[ka];
        a.y = xrow[ka + 1];
        b.x = W[(size_t)ka * D + col0 + l16];
        b.y = W[(size_t)(ka + 1) * D + col0 + l16];
        // (neg_a, A, neg_b, B, c_mod, C, reuse_a, reuse_b)
        c = __builtin_amdgcn_wmma_f32_16x16x4_f32(false, a, false, b,
                                                  (short)0, c, false, false);
    }

    float* hout = H + (size_t)row0 * D + col0 + l16;
#pragma unroll
    for (int v = 0; v < 8; ++v)
        hout[(size_t)(v + 8 * half) * D] = c[v];
}

// ---------------- edge scatter: Agg[dst] += norm * H[src] ---------------------
// One wave per edge; each lane owns a float4 (32 lanes * 4 = 128 feats).
__global__ __launch_bounds__(256)
void k_scatter(const int* __restrict__ ei, int E, int N,
               const float* __restrict__ dinv,
               const float* __restrict__ H, float* __restrict__ Agg) {
    int w    = (int)((blockIdx.x * (unsigned)blockDim.x + threadIdx.x) >> 5);
    int lane = threadIdx.x & 31;
    if (w >= E + N) return;
    int src, dst;
    if (w < E) { src = ei[w]; dst = ei[E + w]; }
    else       { src = w - E; dst = src; }          // self loop
    float norm = dinv[src] * dinv[dst];
    const float4* hs = (const float4*)(H + (size_t)src * D);
    float4 v = hs[lane];
    float* ad = Agg + (size_t)dst * D + lane * 4;
    atomicAdd(ad + 0, v.x * norm);
    atomicAdd(ad + 1, v.y * norm);
    atomicAdd(ad + 2, v.z * norm);
    atomicAdd(ad + 3, v.w * norm);
}

// ---------------- Xn = relu(Agg + b) ------------------------------------------
__global__ void k_bias_relu(const float* __restrict__ Agg, const float* __restrict__ b,
                            float* __restrict__ Xn, int total) {
    int i = blockIdx.x * blockDim.x + threadIdx.x;
    if (i < total) {
        float v = Agg[i] + b[i & (D - 1)];
        Xn[i] = v > 0.f ? v : 0.f;
    }
}

// ---------------- masked segment-sum pool -------------------------------------
__global__ void k_pool(const float* __restrict__ X, const float* __restrict__ mask,
                       const int* __restrict__ batch, float* __restrict__ sums) {
    int n = blockIdx.x;
    int f = threadIdx.x;                 // blockDim.x == D
    float m = mask[n];
    if (m != 0.f)
        atomicAdd(&sums[(size_t)batch[n] * D + f], X[(size_t)n * D + f] * m);
}

// ---------------- out[g] = (sums[g]/max(cnt,1)) . lin_w + lin_b ---------------
__global__ void k_final(const float* __restrict__ sums, const float* __restrict__ cnt,
                        const float* __restrict__ lw, const float* __restrict__ lb,
                        float* __restrict__ out, int G) {
    int g = blockIdx.x * blockDim.x + threadIdx.x;
    if (g >= G) return;
    float c = cnt[g];
    float inv = 1.f / (c > 1.f ? c : 1.f);
    float acc = lb[0];
    for (int f = 0; f < D; ++f) acc += sums[(size_t)g * D + f] * inv * lw[f];
    out[g] = acc;
}

extern "C" void kernel_launch(void* const* d_in, const int* in_sizes, int n_in,
                              void* d_out, int out_size, void* d_ws, size_t ws_size,
                              hipStream_t stream) {
    const float* x     = (const float*)d_in[0];
    const float* Ws    = (const float*)d_in[1];
    const float* bs    = (const float*)d_in[2];
    const float* lw    = (const float*)d_in[3];
    const float* lb    = (const float*)d_in[4];
    const int*   ei    = (const int*)d_in[5];
    const int*   batch = (const int*)d_in[6];

    const int N = in_sizes[0] / D;   // 50000
    const int E = in_sizes[5] / 2;   // 625000
    const int G = out_size;          // 512
    float* out = (float*)d_out;

    // carve workspace
    char* p = (char*)d_ws;
    auto carve = [&](size_t bytes) -> void* {
        void* r = (void*)p; p += (bytes + 255) & ~(size_t)255; return r;
    };
    float* Xbuf = (float*)carve((size_t)N * D * sizeof(float));
    float* H    = (float*)carve((size_t)N * D * sizeof(float));
    float* Agg  = (float*)carve((size_t)N * D * sizeof(float));
    float* deg  = (float*)carve((size_t)N * sizeof(float));
    float* dinv = (float*)carve((size_t)N * sizeof(float));
    float* mask = (float*)carve((size_t)N * sizeof(float));
    float* sums = (float*)carve((size_t)G * D * sizeof(float));
    float* cnt  = (float*)carve((size_t)G * sizeof(float));

    const int T = 256;
    hipMemsetAsync(deg,  0, (size_t)N * sizeof(float), stream);
    hipMemsetAsync(cnt,  0, (size_t)G * sizeof(float), stream);
    hipMemsetAsync(sums, 0, (size_t)G * D * sizeof(float), stream);

    k_deg_edges<<<(E + T - 1) / T, T, 0, stream>>>(ei, E, deg);
    k_dinv<<<(N + T - 1) / T, T, 0, stream>>>(deg, dinv, N);
    k_mask_cnt<<<(N + T - 1) / T, T, 0, stream>>>(x, batch, mask, cnt, N);

    const float* Xcur = x;
    for (int layer = 0; layer < 3; ++layer) {
        k_gemm_wmma_f32<<<N / 16, 256, 0, stream>>>(Xcur, Ws + (size_t)layer * D * D, H);
        hipMemsetAsync(Agg, 0, (size_t)N * D * sizeof(float), stream);
        k_scatter<<<(E + N + 7) / 8, 256, 0, stream>>>(ei, E, N, dinv, H, Agg);
        k_bias_relu<<<((N * D) + T - 1) / T, T, 0, stream>>>(Agg, bs + layer * D, Xbuf, N * D);
        Xcur = Xbuf;  // in-place ping-pong: previous input no longer needed
    }

    k_pool<<<N, D, 0, stream>>>(Xcur, mask, batch, sums);
    k_final<<<(G + T - 1) / T, T, 0, stream>>>(sums, cnt, lw, lb, out, G);
}